// PairLoss_35201551958720
// MI455X (gfx1250) — compile-verified
//
#include <hip/hip_runtime.h>

typedef __attribute__((ext_vector_type(2))) float v2f;
typedef __attribute__((ext_vector_type(8))) float v8f;

#define MARGIN 0.5f

__device__ __forceinline__ float readlane_f(float v, int lane) {
    union { float f; int i; } u;
    u.f = v;
    u.i = __builtin_amdgcn_readlane(u.i, lane);
    return u.f;
}

// Sum x across all 32 lanes of the wave using V_WMMA_F32_16X16X4_F32.
// B = all ones  =>  D[m,n] = sum_k A[m,k]  (row sums of A, replicated per column).
// Per-lane partials fill one A register (other zero): total = sum of all A
// entries = (sum c[0..7] @lane0) + (sum c[0..7] @lane16) per the documented
// 16x16 f32 C/D layout. Result uniform across the wave.
// Requires EXEC == all 1s (call only from wave-uniform control flow).
__device__ __forceinline__ float wave_sum_wmma(float x) {
    v2f a; a[0] = x;    a[1] = 0.0f;
    v2f b; b[0] = 1.0f; b[1] = 1.0f;
    v8f c = {};
    c = __builtin_amdgcn_wmma_f32_16x16x4_f32(
            /*neg_a=*/false, a, /*neg_b=*/false, b,
            /*c_mod=*/(short)0, c, /*reuse_a=*/false, /*reuse_b=*/false);
    float s = ((c[0] + c[1]) + (c[2] + c[3])) + ((c[4] + c[5]) + (c[6] + c[7]));
    return readlane_f(s, 0) + readlane_f(s, 16);
}

// L2 distance between rows i and j of h (256 f32 each), one pair per wave32.
// Lane l reads float4 #l and #(l+32) of each row: two perfectly coalesced
// global_load_b128 per row (full 128B L2 lines; h is L2-resident: 102.4 MB
// vs 192 MB L2). Pair indices are wave-uniform -> readfirstlane them into
// SGPRs so row loads use the saddr + lane*16 addressing form.
__device__ __forceinline__ float pair_dist(const float* __restrict__ h,
                                           const long long* __restrict__ pr,
                                           int lane) {
    const int i = __builtin_amdgcn_readfirstlane((int)pr[0]);
    const int j = __builtin_amdgcn_readfirstlane((int)pr[1]);
    const float4* ri = (const float4*)(h + (size_t)i * 256);
    const float4* rj = (const float4*)(h + (size_t)j * 256);

    float4 a0 = ri[lane];
    float4 a1 = ri[lane + 32];
    float4 b0 = rj[lane];
    float4 b1 = rj[lane + 32];

    float d, p = 0.0f;
    d = a0.x - b0.x; p = fmaf(d, d, p);
    d = a0.y - b0.y; p = fmaf(d, d, p);
    d = a0.z - b0.z; p = fmaf(d, d, p);
    d = a0.w - b0.w; p = fmaf(d, d, p);
    d = a1.x - b1.x; p = fmaf(d, d, p);
    d = a1.y - b1.y; p = fmaf(d, d, p);
    d = a1.z - b1.z; p = fmaf(d, d, p);
    d = a1.w - b1.w; p = fmaf(d, d, p);

    // Single-instruction sqrt (~1 ulp) instead of the multi-op refined one.
    return __builtin_amdgcn_sqrtf(wave_sum_wmma(p));
}

// Two separate uniform grid-stride loops (pos then neg): no per-iteration
// pos/neg select, no exec-mask manipulation inside the hot loop, EXEC stays
// all-1s for the WMMA reductions.
__global__ void PairLoss_kernel(const float* __restrict__ h,
                                const long long* __restrict__ pos,
                                const long long* __restrict__ neg,
                                float* __restrict__ partials,
                                int P) {
    const int tid    = blockIdx.x * blockDim.x + threadIdx.x;
    const int lane   = threadIdx.x & 31;
    const int wave   = tid >> 5;
    const int nwaves = (gridDim.x * blockDim.x) >> 5;

    float loss = 0.0f;
    for (int w = wave; w < P; w += nwaves) {
        loss += pair_dist(h, pos + 2ll * w, lane);
    }
    for (int w = wave; w < P; w += nwaves) {
        const float dist = pair_dist(h, neg + 2ll * w, lane);
        loss += fmaxf(0.0f, MARGIN - dist);
    }
    // loss is uniform across the wave; one write per wave, every slot written.
    if (lane == 0) partials[wave] = loss;
}

// Single-block deterministic final reduction (fixed accumulation order).
__global__ void PairLoss_reduce(const float* __restrict__ partials, int n,
                                float* __restrict__ out) {
    __shared__ float smem[8];
    const int t = threadIdx.x;
    float x = 0.0f;
    for (int k = t; k < n; k += 256) x += partials[k];
    float s = wave_sum_wmma(x);
    if ((t & 31) == 0) smem[t >> 5] = s;
    __syncthreads();
    if (t == 0) {
        float tot = 0.0f;
        for (int k = 0; k < 8; ++k) tot += smem[k];
        out[0] = tot;
    }
}

extern "C" void kernel_launch(void* const* d_in, const int* in_sizes, int n_in,
                              void* d_out, int out_size, void* d_ws, size_t ws_size,
                              hipStream_t stream) {
    const float*     h   = (const float*)d_in[0];
    const long long* pos = (const long long*)d_in[1];
    const long long* neg = (const long long*)d_in[2];
    const int P = in_sizes[1] / 2;

    int blocks = 2048;                       // 16384 waves -> ~25 pairs/wave
    const size_t maxW = ws_size / sizeof(float);
    while ((size_t)(blocks * 8) > maxW && blocks > 1) blocks >>= 1;
    const int W = blocks * 256 / 32;

    float* partials = (float*)d_ws;
    PairLoss_kernel<<<blocks, 256, 0, stream>>>(h, pos, neg, partials, P);
    PairLoss_reduce<<<1, 256, 0, stream>>>(partials, W, (float*)d_out);
}